// SparseCoder_51453708206455
// MI455X (gfx1250) — compile-verified
//
#include <hip/hip_runtime.h>
#include <math.h>

#define K_ORDER   8
#define IN_CH     32
#define NUM_ATOMS 512
#define LAMBDA    0.1f
#define ETA       0.05f
#define NUM_ITERS 50
#define BATCH     16
#define NNODES    128
#define DROWS     (NNODES * IN_CH)   // 4096
#define H_PARTS   8

typedef __attribute__((ext_vector_type(2))) float v2f;
typedef __attribute__((ext_vector_type(8))) float v8f;

#define WMMA_F32(A, B, C) \
    __builtin_amdgcn_wmma_f32_16x16x4_f32(false, (A), false, (B), (short)0, (C), false, false)

// ---------------------------------------------------------------------------
// Kernel 1: per-graph e = eigs/max, powers P[k][m], then T[n,k] = sum_m V[n,m]P[k][m]
// One block per graph, 128 threads. Tiny (1 MB of V total) -> VALU.
// ---------------------------------------------------------------------------
__global__ void build_T_kernel(const float* __restrict__ eigs,
                               const float* __restrict__ V,
                               float* __restrict__ T) {
    __shared__ float smax[NNODES];
    __shared__ float P[K_ORDER][NNODES];
    const int b = blockIdx.x;
    const int m = threadIdx.x;

    float ev = eigs[b * NNODES + m];
    smax[m] = ev;
    __syncthreads();
    for (int s = NNODES / 2; s > 0; s >>= 1) {
        if (m < s) smax[m] = fmaxf(smax[m], smax[m + s]);
        __syncthreads();
    }
    float e = ev / smax[0];
    float p = 1.0f;
    #pragma unroll
    for (int k = 0; k < K_ORDER; ++k) { P[k][m] = p; p *= e; }
    __syncthreads();

    const float* Vrow = V + ((size_t)b * NNODES + m) * NNODES;
    float acc[K_ORDER];
    #pragma unroll
    for (int k = 0; k < K_ORDER; ++k) acc[k] = 0.0f;
    for (int mm = 0; mm < NNODES; ++mm) {
        float v = Vrow[mm];
        #pragma unroll
        for (int k = 0; k < K_ORDER; ++k) acc[k] = fmaf(v, P[k][mm], acc[k]);
    }
    float* Trow = T + ((size_t)b * NNODES + m) * K_ORDER;
    #pragma unroll
    for (int k = 0; k < K_ORDER; ++k) Trow[k] = acc[k];
}

// ---------------------------------------------------------------------------
// Kernel 2: D[b][(n,c), j] = sum_k T[b][n,k] * A[k][c*512 + j].
// Store-bound (134 MB write). idx packs (b,n,c,j) so D writes are the flat index.
// ---------------------------------------------------------------------------
__global__ void build_D_kernel(const float* __restrict__ T,
                               const float* __restrict__ A,
                               float* __restrict__ D) {
    size_t idx = (size_t)blockIdx.x * blockDim.x + threadIdx.x;
    const int j = (int)(idx & (NUM_ATOMS - 1));
    const int c = (int)((idx >> 9) & (IN_CH - 1));
    const int n = (int)((idx >> 14) & (NNODES - 1));
    const int b = (int)(idx >> 21);
    const float* Trow = T + ((size_t)b * NNODES + n) * K_ORDER;
    float acc = 0.0f;
    #pragma unroll
    for (int k = 0; k < K_ORDER; ++k)
        acc = fmaf(Trow[k], A[(size_t)k * (IN_CH * NUM_ATOMS) + c * NUM_ATOMS + j], acc);
    D[idx] = acc;
}

// ---------------------------------------------------------------------------
// Kernel 3: f output = x reshaped (flat copy).
// ---------------------------------------------------------------------------
__global__ void copy_f_kernel(const float* __restrict__ x, float* __restrict__ f) {
    int i = blockIdx.x * blockDim.x + threadIdx.x;
    f[i] = x[i];
}

// ---------------------------------------------------------------------------
// Kernel 4: partial h: hp[b][part][j] = sum_{d in part} D[b][d,j] * f[b][d].
// 8 partials per graph -> 128 blocks streaming D coalesced; summed in FISTA.
// (No float atomics: keeps kernel_launch bitwise deterministic.)
// ---------------------------------------------------------------------------
__global__ void compute_h_kernel(const float* __restrict__ D,
                                 const float* __restrict__ x,
                                 float* __restrict__ hp) {
    const int part = blockIdx.x;           // 0..7
    const int b    = blockIdx.y;
    const int j    = threadIdx.x;          // 0..511
    const int dlen = DROWS / H_PARTS;      // 512
    const float* Db = D + (size_t)b * DROWS * NUM_ATOMS + (size_t)part * dlen * NUM_ATOMS;
    const float* fb = x + (size_t)b * DROWS + part * dlen;
    float acc = 0.0f;
    for (int d = 0; d < dlen; ++d)
        acc = fmaf(Db[(size_t)d * NUM_ATOMS + j], fb[d], acc);
    hp[((size_t)b * H_PARTS + part) * NUM_ATOMS + j] = acc;
}

// ---------------------------------------------------------------------------
// Kernel 5: G[b] = D[b]^T D[b]  via V_WMMA_F32_16X16X4_F32.
// Block = 128 threads (4 waves). Each wave computes a 32(j) x 64(j') strip:
// two A fragments x four B fragments -> 8 WMMAs per 12 fragment loads per
// K-step of 4 (1.5 loads/WMMA). f32 WMMA layout: lane l, vgpr v ->
// (M = l%16, K = v + 2*(l/16)); Gram structure makes A/B loads identical
// and coalesced (lanes 0-15 read consecutive j).
// ---------------------------------------------------------------------------
__global__ void build_G_kernel(const float* __restrict__ D, float* __restrict__ G) {
    const int lane = threadIdx.x & 31;
    const int wave = threadIdx.x >> 5;             // 0..3
    const int b   = blockIdx.z;
    const int j0  = blockIdx.y * 128 + wave * 32;  // row tile of G (2 fragments)
    const int jp0 = blockIdx.x * 64;               // col tile of G (4 fragments)
    const float* Db = D + (size_t)b * DROWS * NUM_ATOMS;

    const int m  = lane & 15;
    const int kh = (lane >> 4) << 1;               // 0 or 2: K pair for this lane half

    v8f acc00 = {}, acc01 = {}, acc02 = {}, acc03 = {};
    v8f acc10 = {}, acc11 = {}, acc12 = {}, acc13 = {};

    #pragma unroll 2
    for (int d = 0; d < DROWS; d += 4) {
        const float* row0 = Db + (size_t)(d + kh) * NUM_ATOMS;
        const float* row1 = row0 + NUM_ATOMS;
        v2f a0, a1, b0, b1, b2, b3;
        a0.x = row0[j0 + m];        a0.y = row1[j0 + m];
        a1.x = row0[j0 + 16 + m];   a1.y = row1[j0 + 16 + m];
        b0.x = row0[jp0 + m];       b0.y = row1[jp0 + m];
        b1.x = row0[jp0 + 16 + m];  b1.y = row1[jp0 + 16 + m];
        b2.x = row0[jp0 + 32 + m];  b2.y = row1[jp0 + 32 + m];
        b3.x = row0[jp0 + 48 + m];  b3.y = row1[jp0 + 48 + m];
        acc00 = WMMA_F32(a0, b0, acc00);
        acc01 = WMMA_F32(a0, b1, acc01);
        acc02 = WMMA_F32(a0, b2, acc02);
        acc03 = WMMA_F32(a0, b3, acc03);
        acc10 = WMMA_F32(a1, b0, acc10);
        acc11 = WMMA_F32(a1, b1, acc11);
        acc12 = WMMA_F32(a1, b2, acc12);
        acc13 = WMMA_F32(a1, b3, acc13);
    }

    // C/D layout: VGPR i -> M = i (lanes 0-15) / M = i+8 (lanes 16-31), N = lane%16
    const int half = lane >> 4;
    #pragma unroll
    for (int i = 0; i < 8; ++i) {
        int r0 = j0 + i + 8 * half;
        float* G0 = G + (size_t)r0 * NUM_ATOMS + jp0 + m;
        G0[0]  = acc00[i];
        G0[16] = acc01[i];
        G0[32] = acc02[i];
        G0[48] = acc03[i];
        float* G1 = G0 + (size_t)16 * NUM_ATOMS;
        G1[0]  = acc10[i];
        G1[16] = acc11[i];
        G1[32] = acc12[i];
        G1[48] = acc13[i];
    }
}

// ---------------------------------------------------------------------------
// Kernel 6: FISTA with g = G z - h. One block (512 threads) per graph,
// z in LDS, G read via the symmetric (column) access for coalescing.
// ---------------------------------------------------------------------------
__global__ void fista_kernel(const float* __restrict__ G,
                             const float* __restrict__ hp,
                             float* __restrict__ r_out) {
    __shared__ float z_s[NUM_ATOMS];
    const int b = blockIdx.x;
    const int j = threadIdx.x;
    const float* Gb  = G  + (size_t)b * NUM_ATOMS * NUM_ATOMS;
    const float* hpb = hp + (size_t)b * H_PARTS * NUM_ATOMS;

    float hj = 0.0f;
    #pragma unroll
    for (int p = 0; p < H_PARTS; ++p) hj += hpb[p * NUM_ATOMS + j];

    float r = 0.0f;
    float t = 1.0f;
    z_s[j] = 0.0f;
    __syncthreads();

    for (int it = 0; it < NUM_ITERS; ++it) {
        float g = 0.0f;
        #pragma unroll 4
        for (int jj = 0; jj < NUM_ATOMS; ++jj)
            g = fmaf(Gb[(size_t)jj * NUM_ATOMS + j], z_s[jj], g);  // G symmetric
        g -= hj;

        float u     = z_s[j] - ETA * g;
        float mag   = fmaxf(fabsf(u) - ETA * LAMBDA, 0.0f);
        float r_new = copysignf(mag, u);
        float t_new = 0.5f * (1.0f + sqrtf(1.0f + 4.0f * t * t));
        float z_new = r_new + ((t - 1.0f) / t_new) * (r_new - r);

        __syncthreads();
        z_s[j] = z_new;
        r = r_new;
        t = t_new;
        __syncthreads();
    }
    r_out[b * NUM_ATOMS + j] = r;
}

// ---------------------------------------------------------------------------
extern "C" void kernel_launch(void* const* d_in, const int* in_sizes, int n_in,
                              void* d_out, int out_size, void* d_ws, size_t ws_size,
                              hipStream_t stream) {
    const float* x    = (const float*)d_in[0];   // [16,128,32]
    const float* eigs = (const float*)d_in[1];   // [16,128]
    const float* V    = (const float*)d_in[2];   // [16,128,128]
    const float* A    = (const float*)d_in[3];   // [8, 32*512]

    float* out   = (float*)d_out;
    float* r_out = out;                                       // 16*512
    float* D     = out + (size_t)BATCH * NUM_ATOMS;           // 16*4096*512
    float* f_out = D + (size_t)BATCH * DROWS * NUM_ATOMS;     // 16*4096

    float* ws = (float*)d_ws;
    float* T  = ws;                                           // 16*128*8
    float* hp = T + (size_t)BATCH * NNODES * K_ORDER;         // 16*8*512
    float* G  = hp + (size_t)BATCH * H_PARTS * NUM_ATOMS;     // 16*512*512

    build_T_kernel<<<BATCH, NNODES, 0, stream>>>(eigs, V, T);

    {
        size_t total = (size_t)BATCH * DROWS * NUM_ATOMS;     // 33.5M
        build_D_kernel<<<(unsigned)(total / 256), 256, 0, stream>>>(T, A, D);
    }

    copy_f_kernel<<<(BATCH * DROWS) / 256, 256, 0, stream>>>(x, f_out);

    {
        dim3 grid(H_PARTS, BATCH);                            // 8 x 16
        compute_h_kernel<<<grid, NUM_ATOMS, 0, stream>>>(D, x, hp);
    }

    {
        dim3 grid(NUM_ATOMS / 64, NUM_ATOMS / 128, BATCH);    // 8 x 4 x 16
        build_G_kernel<<<grid, 128, 0, stream>>>(D, G);
    }

    fista_kernel<<<BATCH, NUM_ATOMS, 0, stream>>>(G, hp, r_out);
}